// RNN_63591285785314
// MI455X (gfx1250) — compile-verified
//
#include <hip/hip_runtime.h>
#include <math.h>

// ---------------------------------------------------------------------------
// RNN forward for MI455X (gfx1250, wave32).
//   Phase 1: Ux = X @ U_w^T + U_b   (32768x1024x1024 fp32 GEMM, WMMA f32 16x16x4)
//   Phase 2: persistent scan kernel, h_t = tanh(Ux_t + h_{t-1} @ W_w^T + W_b),
//            double-buffered async LDS staging + device-wide barrier per step.
// A-tile staging uses GLOBAL_LOAD_ASYNC_TO_LDS_B128 (ASYNCcnt). B tiles are
// staged transposed into a pair-interleaved layout so each WMMA B fragment is
// one aligned 8-byte LDS load (no VGPR repacking).
// Output layout: d_out = [B,S,H] outputs followed by [B,H] h_last (fp32).
// ---------------------------------------------------------------------------

constexpr int BATCH = 64;
constexpr int SEQ   = 512;
constexpr int HID   = 1024;   // OUT_F
constexpr int KDIM  = 1024;   // IN_F

typedef float v2f __attribute__((ext_vector_type(2)));
typedef float v4f __attribute__((ext_vector_type(4)));
typedef float v8f __attribute__((ext_vector_type(8)));

typedef __attribute__((address_space(3))) float lds_float;

static __device__ __forceinline__ v8f wmma_f32_16x16x4(v2f a, v2f b, v8f c) {
  // 8-arg form: (neg_a, A, neg_b, B, c_mod, C, reuse_a, reuse_b)
  return __builtin_amdgcn_wmma_f32_16x16x4_f32(false, a, false, b, (short)0, c,
                                               false, false);
}

// Async copy of 16 contiguous bytes from global address saddr + voff to the LDS address in ldsoff.
static __device__ __forceinline__ void async_copy_b128(unsigned ldsoff,
                                                       int voff,
                                                       const float* saddr) {
  asm volatile("global_load_async_to_lds_b128 %0, %1, %2"
               :: "v"(ldsoff), "v"(voff), "s"(saddr)
               : "memory");
}

static __device__ __forceinline__ void wait_asynccnt0() {
  asm volatile("s_wait_asynccnt 0x0" ::: "memory");
}
static __device__ __forceinline__ void wait_asynccnt2() {
  asm volatile("s_wait_asynccnt 0x2" ::: "memory");
}

static __device__ __forceinline__ unsigned lds_offset(const float* p) {
  return (unsigned)(size_t)(lds_float*)p;
}

// ---------------------------------------------------------------------------
// Phase 1: C[m,n] = sum_k X[m,k] * Uw[n,k] + Ub[n]
// Block tile 64x64, K-chunk 32, 8 waves: wave (wm,wn) computes rows wm*16..+16,
// cols wn*32..+32 (two 16x16 WMMA tiles sharing the A fragment).
// sB pair layout: sB[(k>>1)*128 + 2n + (k&1)]  (row width 64 cols * 2).
// ---------------------------------------------------------------------------
constexpr int KC1 = 32;

__global__ __launch_bounds__(256) void ux_gemm(const float* __restrict__ X,
                                               const float* __restrict__ Uw,
                                               const float* __restrict__ Ub,
                                               float* __restrict__ out) {
  __shared__ float sA[64 * KC1];   // X tile, row-major [64][32]
  __shared__ float sB[KC1 * 64];   // Uw tile, transposed pair-interleaved
  const int m0   = blockIdx.x * 64;
  const int n0   = blockIdx.y * 64;
  const int tid  = threadIdx.x;
  const int lane = tid & 31;
  const int wave = tid >> 5;
  const int half = lane >> 4;      // 0: lanes 0-15, 1: lanes 16-31
  const int lm   = lane & 15;
  const int wm   = wave & 3;       // 16-row sub-tile
  const int wn   = wave >> 2;      // 32-col group

  v8f acc0 = {0.f, 0.f, 0.f, 0.f, 0.f, 0.f, 0.f, 0.f};
  v8f acc1 = acc0;

  for (int k0 = 0; k0 < KDIM; k0 += KC1) {
    __syncthreads();   // protect LDS from previous chunk's consumers
    // Stage A asynchronously: X[m0:m0+64, k0:k0+32] (512 b128 / 256 threads).
    for (int i = tid; i < 64 * (KC1 / 4); i += 256) {
      const int r  = i >> 3;
      const int c4 = (i & 7) << 2;
      async_copy_b128(lds_offset(&sA[r * KC1 + c4]),
                      (int)(((size_t)(m0 + r) * KDIM + k0 + c4) * sizeof(float)),
                      X);
    }
    // Stage B transposed into pair layout: two b64 LDS stores per float4.
    for (int i = tid; i < 64 * (KC1 / 4); i += 256) {
      const int n  = i >> 3;
      const int c4 = (i & 7) << 2;
      v4f v = *(const v4f*)&Uw[(size_t)(n0 + n) * KDIM + k0 + c4];
      *(v2f*)&sB[(c4 >> 1) * 128 + 2 * n]       = (v2f){v.x, v.y};
      *(v2f*)&sB[((c4 >> 1) + 1) * 128 + 2 * n] = (v2f){v.z, v.w};
    }
    // Prefetch next chunk while we compute this one (global_prefetch_b8).
    if (k0 + KC1 < KDIM) {
      __builtin_prefetch(&X[(size_t)(m0 + (tid & 63)) * KDIM + k0 + KC1], 0, 1);
      __builtin_prefetch(&Uw[(size_t)(n0 + (tid & 63)) * KDIM + k0 + KC1], 0, 1);
    }
    wait_asynccnt0();   // this wave's async LDS writes have landed
    __syncthreads();    // ...and so have everyone else's

    const float* As = &sA[(wm * 16 + lm) * KC1 + 2 * half];
    const float* Bs = &sB[half * 128 + 2 * (wn * 32 + lm)];
    #pragma unroll
    for (int kk = 0; kk < KC1; kk += 4) {
      v2f a  = *(const v2f*)(As + kk);
      v2f b0 = *(const v2f*)(Bs + (kk >> 1) * 128);
      v2f b1 = *(const v2f*)(Bs + (kk >> 1) * 128 + 32);
      acc0 = wmma_f32_16x16x4(a, b0, acc0);
      acc1 = wmma_f32_16x16x4(a, b1, acc1);
    }
  }

  // Epilogue: +bias, store. C/D layout: VGPR v -> row (v + 8*half), col lm.
  const int gn = n0 + wn * 32 + lm;
  const float bias0 = Ub[gn];
  const float bias1 = Ub[gn + 16];
  #pragma unroll
  for (int v = 0; v < 8; ++v) {
    const int gm = m0 + wm * 16 + 8 * half + v;
    out[(size_t)gm * HID + gn]      = acc0[v] + bias0;
    out[(size_t)gm * HID + gn + 16] = acc1[v] + bias1;
  }
}

// ---------------------------------------------------------------------------
// Phase 2: persistent recurrent scan. 32 blocks, block b owns columns
// [b*32, b*32+32). Double-buffered K-chunks: async copies for chunk c+1 are
// issued before computing chunk c; s_wait_asynccnt 0x2 retires exactly the
// older chunk's two copies (async loads complete in order per wave).
// ---------------------------------------------------------------------------
constexpr int NBLK2 = HID / 32;   // 32 persistent blocks
constexpr int KC2   = 32;
constexpr int NCH2  = HID / KC2;  // 32 chunks (even -> chunk 0 is buffer 0)

__device__ __forceinline__ void grid_barrier(unsigned* bar) {
  __syncthreads();
  __threadfence();
  if (threadIdx.x == 0) {
    unsigned gen = __hip_atomic_load(&bar[1], __ATOMIC_ACQUIRE,
                                     __HIP_MEMORY_SCOPE_AGENT);
    unsigned arrived = __hip_atomic_fetch_add(&bar[0], 1u, __ATOMIC_ACQ_REL,
                                              __HIP_MEMORY_SCOPE_AGENT);
    if (arrived == (unsigned)(NBLK2 - 1)) {
      __hip_atomic_store(&bar[0], 0u, __ATOMIC_RELAXED,
                         __HIP_MEMORY_SCOPE_AGENT);
      __hip_atomic_fetch_add(&bar[1], 1u, __ATOMIC_RELEASE,
                             __HIP_MEMORY_SCOPE_AGENT);
    } else {
      while (__hip_atomic_load(&bar[1], __ATOMIC_ACQUIRE,
                               __HIP_MEMORY_SCOPE_AGENT) == gen)
        __builtin_amdgcn_s_sleep(2);
    }
  }
  __syncthreads();
}

__global__ __launch_bounds__(256) void rnn_scan(float* __restrict__ out,
                                                const float* __restrict__ Ww,
                                                const float* __restrict__ Wb,
                                                unsigned* __restrict__ bar) {
  __shared__ float sA[2][BATCH * KC2];  // h_{t-1}[0:64, kchunk], double-buffered
  __shared__ float sB[2][KC2 * 32];     // Ww strip chunk, pair-interleaved
  const int n0   = blockIdx.x * 32;
  const int tid  = threadIdx.x;
  const int lane = tid & 31;
  const int wave = tid >> 5;
  const int half = lane >> 4;
  const int lm   = lane & 15;
  const int wm   = wave & 3;    // batch sub-tile (rows wm*16..+16)
  const int wn   = wave >> 2;   // 16-col sub-strip within the 32-col strip
  const int gn   = n0 + wn * 16 + lm;
  const float bias = Wb[gn];

  for (int t = 0; t < SEQ; ++t) {
    // Accumulator starts at Ux[:, t, strip] (written by phase 1).
    v8f acc;
    #pragma unroll
    for (int v = 0; v < 8; ++v) {
      const int m = wm * 16 + 8 * half + v;
      acc[v] = out[(size_t)m * (SEQ * HID) + (size_t)t * HID + gn];
    }

    if (t > 0) {   // h_0 == 0 -> no GEMM at t == 0
      const size_t prev = (size_t)(t - 1) * HID;

      // Stage chunk `ch` into buffer `buf`.
      auto stage = [&](int ch, int buf) {
        const int k0 = ch * KC2;
        // h_{t-1}[0:64, k0:k0+32] via async DMA (2 instr/wave/chunk).
        for (int i = tid; i < BATCH * (KC2 / 4); i += 256) {
          const int r  = i >> 3;
          const int c4 = (i & 7) << 2;
          async_copy_b128(
              lds_offset(&sA[buf][r * KC2 + c4]),
              (int)(((size_t)r * (SEQ * HID) + prev + k0 + c4) * sizeof(float)),
              out);
        }
        // sB pair layout: sB[(k>>1)*64 + 2n + (k&1)], one float4 per thread.
        const int n  = tid >> 3;
        const int c4 = (tid & 7) << 2;
        v4f v = *(const v4f*)&Ww[(size_t)(n0 + n) * HID + k0 + c4];
        *(v2f*)&sB[buf][(c4 >> 1) * 64 + 2 * n]       = (v2f){v.x, v.y};
        *(v2f*)&sB[buf][((c4 >> 1) + 1) * 64 + 2 * n] = (v2f){v.z, v.w};
      };

      stage(0, 0);   // preamble
      for (int c = 0; c < NCH2; ++c) {
        const int cur = c & 1;
        if (c + 1 < NCH2) {
          stage(c + 1, cur ^ 1);
          wait_asynccnt2();   // retire chunk c's two older async copies
        } else {
          wait_asynccnt0();
        }
        __syncthreads();      // chunk c visible block-wide

        const float* As = &sA[cur][(wm * 16 + lm) * KC2 + 2 * half];
        const float* Bs = &sB[cur][half * 64 + 2 * (wn * 16 + lm)];
        #pragma unroll
        for (int kk = 0; kk < KC2; kk += 4) {
          v2f a = *(const v2f*)(As + kk);
          v2f b = *(const v2f*)(Bs + (kk >> 1) * 64);
          acc = wmma_f32_16x16x4(a, b, acc);
        }
        __syncthreads();      // done reading buffer `cur`; safe to restage it
      }
    }

    // tanh + store (overwrite Ux slice t with h_t). Last step also -> h_last.
    #pragma unroll
    for (int v = 0; v < 8; ++v) {
      const int m = wm * 16 + 8 * half + v;
      const float hv = tanhf(acc[v] + bias);
      out[(size_t)m * (SEQ * HID) + (size_t)t * HID + gn] = hv;
      if (t == SEQ - 1)
        out[(size_t)BATCH * SEQ * HID + (size_t)m * HID + gn] = hv;
    }
    grid_barrier(bar);
  }
}

// ---------------------------------------------------------------------------
extern "C" void kernel_launch(void* const* d_in, const int* in_sizes, int n_in,
                              void* d_out, int out_size, void* d_ws,
                              size_t ws_size, hipStream_t stream) {
  (void)in_sizes; (void)n_in; (void)out_size; (void)ws_size;
  const float* X  = (const float*)d_in[0];  // input_emb [B,S,IN]
  const float* Ww = (const float*)d_in[1];  // W_w [H,H]
  const float* Wb = (const float*)d_in[2];  // W_b [H]
  const float* Uw = (const float*)d_in[3];  // U_w [H,IN]
  const float* Ub = (const float*)d_in[4];  // U_b [H]
  float* out = (float*)d_out;
  unsigned* bar = (unsigned*)d_ws;

  // Reset grid-barrier state every call (graph-capture-safe, deterministic).
  hipMemsetAsync(d_ws, 0, 64, stream);

  dim3 g1((BATCH * SEQ) / 64, HID / 64);   // 512 x 16 blocks
  ux_gemm<<<g1, 256, 0, stream>>>(X, Uw, Ub, out);

  rnn_scan<<<dim3(NBLK2), 256, 0, stream>>>(out, Ww, Wb, bar);
}